// attention_48713519071865
// MI455X (gfx1250) — compile-verified
//
#include <hip/hip_runtime.h>
#include <stdint.h>

// Problem constants (reference: B=4, N=4096, D=1024, ctx read from device).
constexpr int B_ = 4;
constexpr int N_ = 4096;
constexpr int D_ = 1024;
constexpr int D2_ = D_ / 2;           // channel *pairs* per row

constexpr int BLOCK_  = 256;          // 8 wave32s; each thread owns 2 channels
constexpr int T_      = 128;          // outputs per thread along N (one tile)
constexpr int RING_   = 8;            // rows per async chunk (double-buffered)
constexpr int NCHUNK_ = T_ / RING_;   // 16 chunks per tile

typedef float v2f __attribute__((ext_vector_type(2)));

// Low 32 bits of a generic (flat) shared pointer == LDS byte offset on CDNA5.
__device__ __forceinline__ uint32_t lds_off(const void* p) {
  return (uint32_t)(uintptr_t)p;
}

__device__ __forceinline__ void async_row_b64(uint32_t lds, const v2f* gaddr) {
  // GLOBAL_LOAD_ASYNC_TO_LDS_B64: vdst = LDS byte-offset VGPR, vaddr = 64-bit address.
  asm volatile("global_load_async_to_lds_b64 %0, %1, off"
               :: "v"(lds), "v"(gaddr) : "memory");
}

__global__ __launch_bounds__(BLOCK_)
void sliding_window_kv_kernel(const float* __restrict__ qf,
                              const float* __restrict__ kf,
                              const float* __restrict__ vf,
                              const int*   __restrict__ ctxp,
                              float*       __restrict__ outf)
{
  // Per-wave private LDS: [wave][buf 0/1][row][k/v][lane] of float2 -> 64 KB.
  __shared__ v2f smem[BLOCK_ / 32][2][RING_][2][32];

  const int ctx  = *ctxp;                                  // 128
  const int b    = blockIdx.z;
  const int pair = blockIdx.y * BLOCK_ + (int)threadIdx.x; // channel-pair index
  const int i0   = blockIdx.x * T_;
  const int wid  = threadIdx.x >> 5;
  const int lane = threadIdx.x & 31;

  const v2f* __restrict__ q2 = (const v2f*)qf;
  const v2f* __restrict__ k2 = (const v2f*)kf;
  const v2f* __restrict__ v2 = (const v2f*)vf;
  v2f*       __restrict__ o2 = (v2f*)outf;

  // element (b, j, pair) = base + j*D2_
  const size_t base = ((size_t)b * N_) * D2_ + (size_t)pair;

  // ---------------- prologue: launch first two async chunks of the leading
  // (j = i + ctx) stream into LDS (ASYNCcnt-tracked, in-order completion).
  const int jaStart = i0 + ctx;
  for (int cc = 0; cc < 2; ++cc) {
    for (int r = 0; r < RING_; ++r) {
      int j  = jaStart + cc * RING_ + r;
      int jc = (j < N_) ? j : (N_ - 1);          // clamp; zero-flagged at use
      async_row_b64(lds_off(&smem[wid][cc][r][0][lane]), k2 + base + (size_t)jc * D2_);
      async_row_b64(lds_off(&smem[wid][cc][r][1][lane]), v2 + base + (size_t)jc * D2_);
    }
  }

  // ---------------- init: W = attn(i0-1) = sum_{j=max(0,i0-ctx-1)}^{i0+ctx-1} relu(k)*v
  v2f W; W.x = 0.0f; W.y = 0.0f;
  {
    int jA = i0 - ctx - 1; if (jA < 0) jA = 0;
    int jB = i0 + ctx - 1; if (jB > N_ - 1) jB = N_ - 1;
    #pragma unroll 4
    for (int j = jA; j <= jB; ++j) {
      v2f kk = k2[base + (size_t)j * D2_];
      v2f vv = v2[base + (size_t)j * D2_];
      W.x += fmaxf(kk.x, 0.0f) * vv.x;
      W.y += fmaxf(kk.y, 0.0f) * vv.y;
    }
  }

  // ---------------- main loop: consume chunk c from LDS, refill with chunk c+2.
  int i = i0;
  for (int c = 0; c < NCHUNK_; ++c) {
    const int buf = c & 1;
    // At most the newest chunk (2*RING_ async loads) may still be in flight.
    asm volatile("s_wait_asynccnt %0" :: "n"(2 * RING_) : "memory");

    #pragma unroll
    for (int r = 0; r < RING_; ++r, ++i) {
      // Leading stream from LDS (landed; ordered by the asynccnt wait above).
      v2f ka = smem[wid][buf][r][0][lane];
      v2f va = smem[wid][buf][r][1][lane];
      int   ja = i + ctx;
      float fa = (ja < N_) ? 1.0f : 0.0f;

      // Lagging stream: 257 rows behind -> L2 hit (whole k+v fits in 192 MB L2).
      int   jr  = i - ctx - 1;
      int   jrc = (jr >= 0) ? jr : 0;
      float fr  = (jr >= 0) ? 1.0f : 0.0f;
      v2f kr = k2[base + (size_t)jrc * D2_];
      v2f vr = v2[base + (size_t)jrc * D2_];

      W.x += fa * (fmaxf(ka.x, 0.0f) * va.x) - fr * (fmaxf(kr.x, 0.0f) * vr.x);
      W.y += fa * (fmaxf(ka.y, 0.0f) * va.y) - fr * (fmaxf(kr.y, 0.0f) * vr.y);

      v2f qv = __builtin_nontemporal_load(q2 + base + (size_t)i * D2_);
      v2f res;
      res.x = fmaxf(qv.x, 0.0f) * W.x;
      res.y = fmaxf(qv.y, 0.0f) * W.y;
      __builtin_nontemporal_store(res, o2 + base + (size_t)i * D2_);
    }

    // Refill the just-consumed buffer with chunk c+2 (clamped; tail chunks are
    // harmless L2-resident reloads; S_ENDPGM's implicit wait-idle drains them).
    const int cc = c + 2;
    for (int r = 0; r < RING_; ++r) {
      int j  = jaStart + cc * RING_ + r;
      int jc = (j < N_) ? j : (N_ - 1);
      async_row_b64(lds_off(&smem[wid][buf][r][0][lane]), k2 + base + (size_t)jc * D2_);
      async_row_b64(lds_off(&smem[wid][buf][r][1][lane]), v2 + base + (size_t)jc * D2_);
    }
  }
}

extern "C" void kernel_launch(void* const* d_in, const int* in_sizes, int n_in,
                              void* d_out, int out_size, void* d_ws, size_t ws_size,
                              hipStream_t stream) {
  const float* q   = (const float*)d_in[0];
  const float* k   = (const float*)d_in[1];
  const float* v   = (const float*)d_in[2];
  const int*   ctx = (const int*)d_in[3];
  float*       out = (float*)d_out;

  // (N/T, D2/BLOCK, B) = (32, 2, 4) = 256 blocks x 8 wave32s = 2048 waves.
  dim3 grid(N_ / T_, D2_ / BLOCK_, B_);
  sliding_window_kv_kernel<<<grid, BLOCK_, 0, stream>>>(q, k, v, ctx, out);
}